// NN_38010460570161
// MI455X (gfx1250) — compile-verified
//
#include <hip/hip_runtime.h>

// out[B,64] = x[B,256] * W^T  (fp32). Memory-bound: ~1.34 GB @ 23.3 TB/s ~ 58 us.
// fp32 WMMA (V_WMMA_F32_16X16X4_F32) + async global->LDS staging of x (ASYNCcnt)
// + W pre-swizzled into B-fragment order in d_ws (WGP$/L2 resident).

typedef float v2f __attribute__((ext_vector_type(2)));
typedef float v8f __attribute__((ext_vector_type(8)));
typedef __attribute__((address_space(3))) float lds_float;

#define K_DIM 256
#define N_DIM 64
#define KC 64               // K columns per staged chunk
#define NCHUNK 4            // 256 / 64
#define ROWS_PER_BLOCK 128  // 8 waves * 16 rows

// ---------------------------------------------------------------------------
// Prep: rearrange W[64][256] into B-fragment-major order in workspace.
// Fragment f = k*4 + n (k = K/4 step, n = N/16 tile), 32 lanes * 8B each:
//   wf[f*64 + lane*2 + e] = W[n*16 + (lane&15)][4k + 2*(lane>>4) + e]
// ---------------------------------------------------------------------------
__global__ void wfrag_prep_kernel(const float* __restrict__ w, float* __restrict__ wf)
{
    const int p = blockIdx.x * 256 + threadIdx.x;  // 0..8191 (one b64 pair each)
    const int l = p & 31;
    const int f = p >> 5;
    const int n = f & 3;
    const int k = f >> 2;
    const int row = n * 16 + (l & 15);
    const int col = 4 * k + 2 * (l >> 4);
    v2f v;
    v.x = w[row * K_DIM + col];
    v.y = w[row * K_DIM + col + 1];
    *(v2f*)(wf + (size_t)p * 2) = v;
}

// ---------------------------------------------------------------------------
// Main GEMM kernel
// ---------------------------------------------------------------------------
__global__ __launch_bounds__(256, 2)
void fc_wmma_async_kernel(const float* __restrict__ x,
                          const float* __restrict__ wf,  // fragment-major W
                          float* __restrict__ out)
{
    // A-fragment-major x staging: lds_x[buf][wave][kstep][frag 256B]
    __shared__ __align__(16) float lds_x[2 * 8192];  // 2 x 32 KB double buffer

    const int tid  = threadIdx.x;
    const int wave = tid >> 5;
    const int lane = tid & 31;
    const int l16  = lane & 15;
    const int half = lane >> 4;

    const int mblk = blockIdx.x * ROWS_PER_BLOCK;
    const int m0   = mblk + wave * 16;

    // ---- staging geometry: thread (r0, kq) copies 16B of row (mblk+r0+16i) ----
    const int r0 = tid >> 4;   // row within 16-row group
    const int kq = tid & 15;   // 16-byte column quad within chunk
    const float* gbase = x + (size_t)(mblk + r0) * K_DIM + kq * 4;
    const unsigned lbase = (unsigned)(unsigned long long)(lds_float*)lds_x
                         + (unsigned)(kq * 256 + r0 * 16);

    v8f acc0 = {}, acc1 = {}, acc2 = {}, acc3 = {};

    // A-fragment read base (lane-linear within 256B fragment, conflict-free)
    const float* ap0 = lds_x + wave * 1024 + l16 * 4 + half * 2;
    // B-fragment read base (lane-linear, 256B contiguous per fragment)
    const float* bp  = wf + lane * 2;

    // async-stage one 128-row x 64-col chunk (8 x b128 per thread)
    auto stage = [&](int chunk, int buf) {
        const float*   g  = gbase + chunk * KC;
        const unsigned d0 = lbase + (unsigned)buf * 32768u;
        #pragma unroll
        for (int i = 0; i < 8; ++i) {
            const float*   gp = g + i * (16 * K_DIM);
            const unsigned dp = d0 + (unsigned)(i * 4096);
            asm volatile("global_load_async_to_lds_b128 %0, %1, off"
                         :: "v"(dp), "v"(gp)
                         : "memory");
        }
    };

    stage(0, 0);  // prologue: chunk 0 -> buffer 0

    for (int c = 0; c < NCHUNK; ++c) {
        if (c + 1 < NCHUNK) {
            stage(c + 1, (c + 1) & 1);                       // prefetch next chunk
            asm volatile("s_wait_asynccnt 0x8" ::: "memory"); // drain chunk c only
        } else {
            asm volatile("s_wait_asynccnt 0x0" ::: "memory");
        }
        __syncthreads();  // chunk c visible to all waves

        const float* ap = ap0 + (c & 1) * 8192;
        const float* bc = bp + c * (16 * 256);

        #pragma unroll
        for (int kk = 0; kk < 16; ++kk) {
            const v2f a  = *(const v2f*)(ap + kk * 64);
            const v2f b0 = *(const v2f*)(bc + kk * 256 + 0);
            const v2f b1 = *(const v2f*)(bc + kk * 256 + 64);
            const v2f b2 = *(const v2f*)(bc + kk * 256 + 128);
            const v2f b3 = *(const v2f*)(bc + kk * 256 + 192);

            acc0 = __builtin_amdgcn_wmma_f32_16x16x4_f32(false, a, false, b0, (short)0, acc0, false, false);
            acc1 = __builtin_amdgcn_wmma_f32_16x16x4_f32(false, a, false, b1, (short)0, acc1, false, false);
            acc2 = __builtin_amdgcn_wmma_f32_16x16x4_f32(false, a, false, b2, (short)0, acc2, false, false);
            acc3 = __builtin_amdgcn_wmma_f32_16x16x4_f32(false, a, false, b3, (short)0, acc3, false, false);
        }

        __syncthreads();  // all waves done reading buf[c&1] before it is re-staged
    }

    // C/D layout: VGPR e, lanes 0-15 -> (M=e, N=lane); lanes 16-31 -> (M=e+8, N=lane-16)
    #pragma unroll
    for (int e = 0; e < 8; ++e) {
        const int row = m0 + e + half * 8;
        float* orow = out + (size_t)row * N_DIM + l16;
        orow[0]  = acc0[e];
        orow[16] = acc1[e];
        orow[32] = acc2[e];
        orow[48] = acc3[e];
    }
}

extern "C" void kernel_launch(void* const* d_in, const int* in_sizes, int n_in,
                              void* d_out, int out_size, void* d_ws, size_t ws_size,
                              hipStream_t stream) {
    const float* x = (const float*)d_in[0];   // [B, 256]
    const float* w = (const float*)d_in[1];   // [64, 256]
    float* out = (float*)d_out;               // [B, 64]
    float* wf  = (float*)d_ws;                // 64 KB fragment-major W

    wfrag_prep_kernel<<<32, 256, 0, stream>>>(w, wf);

    const int batch = in_sizes[0] / K_DIM;            // 1048576
    const int grid  = batch / ROWS_PER_BLOCK;         // 8192
    fc_wmma_async_kernel<<<grid, 256, 0, stream>>>(x, wf, out);
}